// AttentionModel_1494648619255
// MI455X (gfx1250) — compile-verified
//
#include <hip/hip_runtime.h>

typedef __attribute__((ext_vector_type(2))) float v2f;
typedef __attribute__((ext_vector_type(8))) float v8f;

#define BATCH   32
#define SEQ     2048
#define DIM     512
#define NCLS    2
#define THREADS 256
#define WAVES   8

__global__ __launch_bounds__(THREADS)
void attn_pool_cls_kernel(const int*   __restrict__ tokens,
                          const float* __restrict__ emb,
                          const float* __restrict__ cls_w,
                          const float* __restrict__ cls_b,
                          float*       __restrict__ out)
{
    __shared__ float qv[DIM];            // query row; later reused as pooled vector
    __shared__ float sc[SEQ];            // scores -> softmax weights
    __shared__ float part[WAVES * DIM];  // per-wave weighted-sum partials
    __shared__ float red[2 * THREADS];   // block reductions

    const int b    = blockIdx.x;
    const int tid  = threadIdx.x;
    const int lane = tid & 31;
    const int wave = tid >> 5;

    const int* tok = tokens + b * SEQ;

    // ---- stage 0: q = emb[tok[0]] into LDS ----
    {
        const float* qrow = emb + (size_t)tok[0] * DIM;
        qv[2 * tid]     = qrow[2 * tid];
        qv[2 * tid + 1] = qrow[2 * tid + 1];
    }
    __syncthreads();

    // ---- stage 1: scores via chained V_WMMA_F32_16X16X4_F32 (exact f32) ----
    // A (16x4): lanes 0-15 hold K=0,1 ; lanes 16-31 hold K=2,3  -> 8B load/lane
    // B (4x16): q replicated across all N columns (same half-wave K mapping)
    const int koff = (lane >> 4) * 2;      // K sub-offset for this half-wave
    for (int g = wave; g < SEQ / 16; g += WAVES) {
        const int t0  = g * 16;
        const int row = lane & 15;
        const float* krow = emb + (size_t)tok[t0 + row] * DIM;
        v8f c = {};
        #pragma unroll 8
        for (int kk = 0; kk < DIM; kk += 4) {
            v2f a, q2;
            a.x  = krow[kk + koff];
            a.y  = krow[kk + koff + 1];
            q2.x = qv[kk + koff];
            q2.y = qv[kk + koff + 1];
            c = __builtin_amdgcn_wmma_f32_16x16x4_f32(
                    /*neg_a=*/false, a, /*neg_b=*/false, q2,
                    /*c_mod=*/(short)0, c, /*reuse_a=*/false, /*reuse_b=*/false);
        }
        // D layout: lanes 0-15 hold rows M=0..7 in c[0..7]; lanes 16-31 hold M=8..15
        if ((lane & 15) == 0) {
            const int mbase = t0 + (lane >> 4) * 8;
            sc[mbase + 0] = c[0]; sc[mbase + 1] = c[1];
            sc[mbase + 2] = c[2]; sc[mbase + 3] = c[3];
            sc[mbase + 4] = c[4]; sc[mbase + 5] = c[5];
            sc[mbase + 6] = c[6]; sc[mbase + 7] = c[7];
        }
    }
    __syncthreads();

    // ---- stage 2: softmax over sc[0..SEQ) ----
    float m = -3.4e38f;
    for (int t = tid; t < SEQ; t += THREADS) m = fmaxf(m, sc[t]);
    red[tid] = m;
    __syncthreads();
    for (int s = THREADS / 2; s > 0; s >>= 1) {
        if (tid < s) red[tid] = fmaxf(red[tid], red[tid + s]);
        __syncthreads();
    }
    m = red[0];
    __syncthreads();

    float lsum = 0.f;
    for (int t = tid; t < SEQ; t += THREADS) {
        const float w = __expf(sc[t] - m);
        sc[t] = w;
        lsum += w;
    }
    red[tid] = lsum;
    __syncthreads();
    for (int s = THREADS / 2; s > 0; s >>= 1) {
        if (tid < s) red[tid] += red[tid + s];
        __syncthreads();
    }
    const float inv_l = 1.0f / red[0];
    __syncthreads();

    // ---- stage 3: pooled = (1/l) * sum_t w_t * emb[tok[t]] ----
    // wave w owns t in [256w, 256w+256); float4 coalesced row reads (L2 hits)
    float4 acc[4] = {};
    const int t_lo = wave * (SEQ / WAVES);
    for (int t = t_lo; t < t_lo + SEQ / WAVES; ++t) {
        const float wt = sc[t];
        const float4* krow = (const float4*)(emb + (size_t)tok[t] * DIM);
        #pragma unroll
        for (int i = 0; i < 4; ++i) {
            const float4 xv = krow[i * 32 + lane];
            acc[i].x += wt * xv.x;  acc[i].y += wt * xv.y;
            acc[i].z += wt * xv.z;  acc[i].w += wt * xv.w;
        }
    }
    #pragma unroll
    for (int i = 0; i < 4; ++i) {
        const int d0 = i * 128 + lane * 4;
        part[wave * DIM + d0 + 0] = acc[i].x;
        part[wave * DIM + d0 + 1] = acc[i].y;
        part[wave * DIM + d0 + 2] = acc[i].z;
        part[wave * DIM + d0 + 3] = acc[i].w;
    }
    __syncthreads();

    {   // combine wave partials -> pooled (reuse qv)
        float p0 = 0.f, p1 = 0.f;
        #pragma unroll
        for (int w = 0; w < WAVES; ++w) {
            p0 += part[w * DIM + 2 * tid];
            p1 += part[w * DIM + 2 * tid + 1];
        }
        qv[2 * tid]     = p0 * inv_l;
        qv[2 * tid + 1] = p1 * inv_l;
    }
    __syncthreads();

    // ---- stage 4: logits[b][c] = pooled . cls_w[c] + cls_b[c] ----
    {
        const float p0 = qv[2 * tid], p1 = qv[2 * tid + 1];
        red[tid]           = p0 * cls_w[0 * DIM + 2 * tid] + p1 * cls_w[0 * DIM + 2 * tid + 1];
        red[THREADS + tid] = p0 * cls_w[1 * DIM + 2 * tid] + p1 * cls_w[1 * DIM + 2 * tid + 1];
    }
    __syncthreads();
    for (int s = THREADS / 2; s > 0; s >>= 1) {
        if (tid < s) {
            red[tid]           += red[tid + s];
            red[THREADS + tid] += red[THREADS + tid + s];
        }
        __syncthreads();
    }
    if (tid == 0) {
        out[b * NCLS + 0] = red[0]       + cls_b[0];
        out[b * NCLS + 1] = red[THREADS] + cls_b[1];
    }
}

extern "C" void kernel_launch(void* const* d_in, const int* in_sizes, int n_in,
                              void* d_out, int out_size, void* d_ws, size_t ws_size,
                              hipStream_t stream) {
    (void)in_sizes; (void)n_in; (void)out_size; (void)d_ws; (void)ws_size;
    const int*   tokens = (const int*)  d_in[0];   // (32, 2048)
    const float* emb    = (const float*)d_in[1];   // (32000, 512)
    const float* cls_w  = (const float*)d_in[2];   // (2, 512)
    const float* cls_b  = (const float*)d_in[3];   // (2,)
    float* out = (float*)d_out;                    // (32, 2)

    attn_pool_cls_kernel<<<BATCH, THREADS, 0, stream>>>(tokens, emb, cls_w, cls_b, out);
}